// ISI_22522808500306
// MI455X (gfx1250) — compile-verified
//
#include <hip/hip_runtime.h>

// ---------------------------------------------------------------------------
// out[b,t,c] = x[b,t,c] + sum_{s=max(0,t-8)}^{t-1} isi[b,s,c]
// where isi = complex(h)*complex(x) over channel pairs (c, c+64).
// Pure HBM-streaming problem (384 MB @ 23.3 TB/s ~= 16.5 us); optimized for
// coalesced wave32 b64 traffic + CDNA5 async global->LDS double buffering.
// ---------------------------------------------------------------------------

#define B_     32
#define T_     8192
#define C_     128
#define F_     64
#define W_     8
#define CHUNK  128                    // timesteps per wave slice
#define TILE   8                      // timesteps staged per LDS tile
#define NTILES (CHUNK / TILE)         // 16
#define WAVES  8                      // waves per block (256 threads)
#define SLICES (B_ * (T_ / CHUNK))    // 32 * 64 = 2048

typedef float v2f __attribute__((ext_vector_type(2)));
typedef int   v4i __attribute__((vector_size(16)));

#ifndef __has_builtin
#define __has_builtin(x) 0
#endif

#if defined(__HIP_DEVICE_COMPILE__) && defined(__gfx1250__) && \
    __has_builtin(__builtin_amdgcn_global_load_async_to_lds_b128) && \
    __has_builtin(__builtin_amdgcn_s_wait_asynccnt)
#define USE_ASYNC 1
#else
#define USE_ASYNC 0
#endif

__device__ __forceinline__ v2f ld2(const float* p) { return *(const v2f*)p; }

#if USE_ASYNC
// Exact pointer types the builtin wants: (int4 AS1*, int4 AS3*, imm, imm).
typedef __attribute__((address_space(1))) v4i* gv4p;
typedef __attribute__((address_space(3))) v4i* lv4p;

// Copy one 4096-byte tile (TILE rows x 128 ch x f32) global -> LDS with
// 8 per-lane async b128 transfers (32 lanes * 16B = 512B per instruction).
__device__ __forceinline__ void stage_tile(const float* gsrc, float* lds_dst, int lane) {
#pragma unroll
  for (int k = 0; k < 8; ++k) {
    const int off = (k * 32 + lane) * 16;
    __builtin_amdgcn_global_load_async_to_lds_b128(
        (gv4p)((char*)gsrc + off),
        (lv4p)((char*)lds_dst + off),
        0, 0);
  }
}
#endif

__global__ __launch_bounds__(256) void isi_win_kernel(
    const float* __restrict__ x, const float* __restrict__ h,
    float* __restrict__ out)
{
  const int lane = threadIdx.x & 31;
  const int wave = threadIdx.x >> 5;
  const int slice = blockIdx.x * WAVES + wave;          // [0, SLICES)
  const int chunksPerB = T_ / CHUNK;
  const int b  = slice / chunksPerB;
  const int t0 = (slice % chunksPerB) * CHUNK;
  const size_t base = (size_t)b * T_ * C_;
  const float* __restrict__ xb = x + base;
  const float* __restrict__ hb = h + base;
  float*       __restrict__ ob = out + base;
  const int c0 = lane * 2;                              // channel pair columns

  // 8-slot ring of isi values; slot = t & 7 (t0 is a multiple of 8).
  v2f ring_r[W_], ring_i[W_];
#pragma unroll
  for (int k = 0; k < W_; ++k) {
    ring_r[k] = (v2f)0.0f;
    ring_i[k] = (v2f)0.0f;
  }

  // Halo priming: isi for s in [t0-8, t0). t0==0 -> window starts empty.
  if (t0 > 0) {
#pragma unroll
    for (int k = 0; k < W_; ++k) {
      const size_t r = (size_t)(t0 - W_ + k) * C_;
      v2f xr = ld2(xb + r + c0);
      v2f xi = ld2(xb + r + c0 + F_);
      v2f hr = ld2(hb + r + c0);
      v2f hi = ld2(hb + r + c0 + F_);
      ring_r[k] = hr * xr - hi * xi;
      ring_i[k] = hr * xi + hi * xr;
    }
  }

#if USE_ASYNC
  // [wave][x/h][double-buffer][TILE*C_] = 128 KB (fits 320 KB/WGP twice).
  __shared__ float lds[WAVES][2][2][TILE * C_];
  stage_tile(xb + (size_t)t0 * C_, &lds[wave][0][0][0], lane);
  stage_tile(hb + (size_t)t0 * C_, &lds[wave][1][0][0], lane);
#endif

  for (int it = 0; it < NTILES; ++it) {
    const int tt = t0 + it * TILE;

#if USE_ASYNC
    const int buf = it & 1;
    if (it + 1 < NTILES) {
      stage_tile(xb + (size_t)(tt + TILE) * C_, &lds[wave][0][buf ^ 1][0], lane);
      stage_tile(hb + (size_t)(tt + TILE) * C_, &lds[wave][1][buf ^ 1][0], lane);
      __builtin_amdgcn_s_wait_asynccnt(16);   // current tile's 16 loads done
    } else {
      __builtin_amdgcn_s_wait_asynccnt(0);
    }
    const float* lx = &lds[wave][0][buf][0];
    const float* lh = &lds[wave][1][buf][0];
#else
    if (it + 1 < NTILES) {
      // gfx1250 global_prefetch_b8: one 128B line per lane covers next tile.
      __builtin_prefetch(xb + (size_t)(tt + TILE) * C_ + lane * 32, 0, 0);
      __builtin_prefetch(hb + (size_t)(tt + TILE) * C_ + lane * 32, 0, 0);
    }
#endif

#pragma unroll
    for (int j = 0; j < TILE; ++j) {          // ring slot == j (tt % 8 == 0)
      const size_t r = (size_t)(tt + j) * C_;
#if USE_ASYNC
      v2f xr = ld2(lx + j * C_ + c0);
      v2f xi = ld2(lx + j * C_ + c0 + F_);
      v2f hr = ld2(lh + j * C_ + c0);
      v2f hi = ld2(lh + j * C_ + c0 + F_);
#else
      v2f xr = ld2(xb + r + c0);
      v2f xi = ld2(xb + r + c0 + F_);
      v2f hr = ld2(hb + r + c0);
      v2f hi = ld2(hb + r + c0 + F_);
#endif
      // Exclusive window sum: full 8-term tree (no incremental drift).
      v2f s_r = ((ring_r[0] + ring_r[1]) + (ring_r[2] + ring_r[3])) +
                ((ring_r[4] + ring_r[5]) + (ring_r[6] + ring_r[7]));
      v2f s_i = ((ring_i[0] + ring_i[1]) + (ring_i[2] + ring_i[3])) +
                ((ring_i[4] + ring_i[5]) + (ring_i[6] + ring_i[7]));
      v2f o_r = xr + s_r;
      v2f o_i = xi + s_i;
      __builtin_nontemporal_store(o_r, (v2f*)(ob + r + c0));
      __builtin_nontemporal_store(o_i, (v2f*)(ob + r + c0 + F_));
      // Insert isi[t] after emitting out[t] (window excludes current step).
      ring_r[j] = hr * xr - hi * xi;
      ring_i[j] = hr * xi + hi * xr;
    }
  }
}

extern "C" void kernel_launch(void* const* d_in, const int* in_sizes, int n_in,
                              void* d_out, int out_size, void* d_ws, size_t ws_size,
                              hipStream_t stream) {
  (void)in_sizes; (void)n_in; (void)out_size; (void)d_ws; (void)ws_size;
  const float* x = (const float*)d_in[0];
  const float* h = (const float*)d_in[1];
  float* out = (float*)d_out;
  dim3 grid(SLICES / WAVES);   // 256 blocks
  dim3 block(WAVES * 32);      // 256 threads = 8 wave32
  isi_win_kernel<<<grid, block, 0, stream>>>(x, h, out);
}